// OIMLossIoU_9105330668000
// MI455X (gfx1250) — compile-verified
//
#include <hip/hip_runtime.h>
#include <math.h>
#include <stdint.h>

typedef __attribute__((ext_vector_type(2))) float v2f;
typedef __attribute__((ext_vector_type(8))) float v8f;

#define NF      256
#define NP      100000
#define NQ      32768
#define MTOT    (NP + NQ)      // 132768
#define NROWS   1024
#define OIMS    30.0f
#define EPSN    1e-12f
#define NEGBIG  -3.4028235e38f

// GEMM tiling
#define KC      32             // K chunk staged per buffer
#define NCH     (NF / KC)      // 8 chunks
#define PADK    36             // padded LDS k-stride (conflict-free b64 reads)
#define NBM_OUT 260            // ceil(132768 / (4*128)) M partial blocks

// ---------------------------------------------------------------------------
// K0: L2-normalize rows of inputs -> xn
// ---------------------------------------------------------------------------
__global__ __launch_bounds__(256) void normalize_rows(const float* __restrict__ in,
                                                      float* __restrict__ out) {
    __shared__ float red[8];
    const int row = blockIdx.x, t = threadIdx.x;
    float v = in[(size_t)row * NF + t];
    float ss = v * v;
    ss += __shfl_xor(ss, 1, 32);
    ss += __shfl_xor(ss, 2, 32);
    ss += __shfl_xor(ss, 4, 32);
    ss += __shfl_xor(ss, 8, 32);
    ss += __shfl_xor(ss, 16, 32);
    if ((t & 31) == 0) red[t >> 5] = ss;
    __syncthreads();
    float tot = 0.f;
#pragma unroll
    for (int k = 0; k < 8; ++k) tot += red[k];
    float n = sqrtf(tot);
    out[(size_t)row * NF + t] = v / fmaxf(n, EPSN);
}

// ---------------------------------------------------------------------------
// K1: fused GEMM (xn @ [lut;cq]^T) + masking + online-softmax partials.
// Block: 256 thr (8 waves). Tile: 64 rows x 128 cols, 4 M-subtiles.
// K streamed in 8 chunks of 32 through DOUBLE-BUFFERED LDS filled with
// GLOBAL_LOAD_ASYNC_TO_LDS_B128 (ASYNCcnt-pipelined). WMMA f32 16x16x4.
// ---------------------------------------------------------------------------
__global__ __launch_bounds__(256) void oim_gemm_softmax(
    const float* __restrict__ xn, const float* __restrict__ lut,
    const float* __restrict__ cq, float* __restrict__ partM,
    float* __restrict__ partS) {
    __shared__ float ldsX[2][64][PADK];    // 18,432 B
    __shared__ float ldsW[2][128][PADK];   // 36,864 B
    __shared__ float pm[64][8], ps[64][8]; //  4,096 B
    __shared__ float rowM[64], rowS[64];   //    512 B

    const int t    = threadIdx.x;
    const int w    = t >> 5;        // wave 0..7 -> 16-col chunk
    const int lane = t & 31;
    const int lo   = lane & 15;
    const int h    = lane >> 4;     // lane half
    const int mbO  = blockIdx.x;    // 0..259
    const int r0   = blockIdx.y * 64;

    // per-thread staging coordinates (16B per lane per issue)
    const int xrow0 = t >> 3,        xseg0 = (t & 7) * 4;         // j = 0
    const int xrow1 = (t + 256) >> 3, xseg1 = ((t + 256) & 7) * 4; // j = 1

    if (t < 64) { rowM[t] = NEGBIG; rowS[t] = 0.f; }
    __syncthreads();

    for (int it = 0; it < 4; ++it) {
        const int c0 = (mbO * 4 + it) * 128;
        if (c0 >= MTOT) break;                       // block-uniform
        v8f acc[4] = {};
        int nzf = 0;

        // issue one K-chunk (6 async b128 instructions per wave)
        auto stage = [&](int buf, int kc) {
            const int k0 = kc * KC;
            {   // xn tile: 64 rows x 32 k  (2 issues / thread)
                unsigned la0 = (unsigned)(uintptr_t)&ldsX[buf][xrow0][xseg0];
                unsigned long long ga0 =
                    (unsigned long long)(uintptr_t)&xn[(size_t)(r0 + xrow0) * NF + k0 + xseg0];
                asm volatile("global_load_async_to_lds_b128 %0, %1, off"
                             :: "v"(la0), "v"(ga0) : "memory");
                unsigned la1 = (unsigned)(uintptr_t)&ldsX[buf][xrow1][xseg1];
                unsigned long long ga1 =
                    (unsigned long long)(uintptr_t)&xn[(size_t)(r0 + xrow1) * NF + k0 + xseg1];
                asm volatile("global_load_async_to_lds_b128 %0, %1, off"
                             :: "v"(la1), "v"(ga1) : "memory");
            }
            // bank tile: 128 cols x 32 k (4 issues / thread), OOB clamps to
            // the (all-zero) last cq row; those columns are masked later.
#pragma unroll
            for (int j = 0; j < 4; ++j) {
                int idx = t + 256 * j;
                int col = idx >> 3, seg = (idx & 7) * 4;
                int gcol = c0 + col;
                int gc = (gcol < MTOT) ? gcol : (MTOT - 1);
                const float* src = (gc < NP) ? (lut + (size_t)gc * NF)
                                             : (cq + (size_t)(gc - NP) * NF);
                unsigned la = (unsigned)(uintptr_t)&ldsW[buf][col][seg];
                unsigned long long ga =
                    (unsigned long long)(uintptr_t)&src[k0 + seg];
                asm volatile("global_load_async_to_lds_b128 %0, %1, off"
                             :: "v"(la), "v"(ga) : "memory");
            }
        };

        stage(0, 0);                                  // prologue
        for (int kc = 0; kc < NCH; ++kc) {
            const int cur = kc & 1;
            if (kc + 1 < NCH) {
                stage(cur ^ 1, kc + 1);               // prefetch next chunk
                asm volatile("s_wait_asynccnt 6" ::: "memory");
            } else {
                asm volatile("s_wait_asynccnt 0" ::: "memory");
            }
            __syncthreads();                          // chunk `cur` visible to all
            const int cb = w * 16;
            for (int kk = 0; kk < KC; kk += 4) {
                v2f b = *(const v2f*)&ldsW[cur][cb + lo][kk + 2 * h];
                nzf |= (b[0] != 0.f) | (b[1] != 0.f); // fused bad-column probe
#pragma unroll
                for (int rt = 0; rt < 4; ++rt) {
                    v2f a = *(const v2f*)&ldsX[cur][rt * 16 + lo][kk + 2 * h];
                    acc[rt] = __builtin_amdgcn_wmma_f32_16x16x4_f32(
                        false, a, false, b, (short)0, acc[rt], false, false);
                }
            }
            __syncthreads();                          // done reading `cur`
        }
        // column nonzero flag: lane pair (lo, lo+16) together covers all K
        int nzAll = nzf | __shfl_xor(nzf, 16, 32);
        const int  gcol  = c0 + w * 16 + lo;
        const bool valid = gcol < MTOT;
        const bool bad   = valid && (nzAll == 0);
#pragma unroll
        for (int rt = 0; rt < 4; ++rt) {
#pragma unroll
            for (int i = 0; i < 8; ++i) {
                float d  = acc[rt][i];
                float zv = bad ? -1.f : d;
                float z  = valid ? OIMS * zv : NEGBIG;
                float m  = z;
                m = fmaxf(m, __shfl_xor(m, 1, 32));
                m = fmaxf(m, __shfl_xor(m, 2, 32));
                m = fmaxf(m, __shfl_xor(m, 4, 32));
                m = fmaxf(m, __shfl_xor(m, 8, 32));
                float e = valid ? __expf(z - m) : 0.f;
                float s = e;
                s += __shfl_xor(s, 1, 32);
                s += __shfl_xor(s, 2, 32);
                s += __shfl_xor(s, 4, 32);
                s += __shfl_xor(s, 8, 32);
                if (lo == 0) {                 // lanes 0 & 16 (C layout: M = i + 8h)
                    int rl = rt * 16 + i + 8 * h;
                    pm[rl][w] = m;
                    ps[rl][w] = s;
                }
            }
        }
        __syncthreads();
        if (t < 64) {                          // combine 8 waves, merge into carry
            float M = NEGBIG;
#pragma unroll
            for (int ww = 0; ww < 8; ++ww) M = fmaxf(M, pm[t][ww]);
            float S = 0.f;
#pragma unroll
            for (int ww = 0; ww < 8; ++ww) {
                float sw = ps[t][ww];
                if (sw > 0.f) S += sw * __expf(pm[t][ww] - M);
            }
            float Mn = fmaxf(rowM[t], M);
            float Sa = (rowS[t] > 0.f ? rowS[t] * __expf(rowM[t] - Mn) : 0.f) +
                       (S > 0.f ? S * __expf(M - Mn) : 0.f);
            rowM[t] = Mn;
            rowS[t] = Sa;
        }
        __syncthreads();
    }
    if (t < 64) {
        partM[(size_t)(r0 + t) * NBM_OUT + mbO] = rowM[t];
        partS[(size_t)(r0 + t) * NBM_OUT + mbO] = rowS[t];
    }
}

// ---------------------------------------------------------------------------
// K2: per-row label logit + lut-row bad flag (labeled rows only)
// ---------------------------------------------------------------------------
__global__ __launch_bounds__(256) void label_logit(const int* __restrict__ label,
                                                   const float* __restrict__ xn,
                                                   const float* __restrict__ lut,
                                                   float* __restrict__ labVal,
                                                   int* __restrict__ labBad) {
    __shared__ float redD[8], redZ[8];
    const int i = blockIdx.x, t = threadIdx.x;
    const int y = label[i];
    if (y >= NP) {
        if (t == 0) { labVal[i] = 0.f; labBad[i] = 0; }
        return;
    }
    float l = lut[(size_t)y * NF + t];
    float x = xn[(size_t)i * NF + t];
    float d = l * x;
    float nz = (l != 0.f) ? 1.f : 0.f;
#pragma unroll
    for (int m = 1; m <= 16; m <<= 1) {
        d  += __shfl_xor(d, m, 32);
        nz += __shfl_xor(nz, m, 32);
    }
    if ((t & 31) == 0) { redD[t >> 5] = d; redZ[t >> 5] = nz; }
    __syncthreads();
    if (t == 0) {
        float D = 0.f, Z = 0.f;
#pragma unroll
        for (int k = 0; k < 8; ++k) { D += redD[k]; Z += redZ[k]; }
        int bad = (Z == 0.f) ? 1 : 0;
        labBad[i] = bad;
        labVal[i] = bad ? -1.f : D;
    }
}

// ---------------------------------------------------------------------------
// K3: reduce 260 partials per row -> per-row iou^2 * CE (with bad_pos fixup)
// ---------------------------------------------------------------------------
__global__ __launch_bounds__(256) void row_finalize(
    const float* __restrict__ partM, const float* __restrict__ partS,
    const int* __restrict__ label, const float* __restrict__ iou,
    const float* __restrict__ labVal, const int* __restrict__ labBad,
    float* __restrict__ rowLoss) {
    __shared__ float sm[256], ssum[256];
    const int i = blockIdx.x, t = threadIdx.x;
    float m = NEGBIG;
    for (int j = t; j < NBM_OUT; j += 256)
        m = fmaxf(m, partM[(size_t)i * NBM_OUT + j]);
    sm[t] = m;
    __syncthreads();
    for (int st = 128; st > 0; st >>= 1) {
        if (t < st) sm[t] = fmaxf(sm[t], sm[t + st]);
        __syncthreads();
    }
    const float M = sm[0];
    float s = 0.f;
    for (int j = t; j < NBM_OUT; j += 256) {
        float sw = partS[(size_t)i * NBM_OUT + j];
        if (sw > 0.f) s += sw * __expf(partM[(size_t)i * NBM_OUT + j] - M);
    }
    ssum[t] = s;
    __syncthreads();
    for (int st = 128; st > 0; st >>= 1) {
        if (t < st) ssum[t] += ssum[t + st];
        __syncthreads();
    }
    if (t == 0) {
        float S = ssum[0];
        float out = 0.f;
        int y = label[i];
        if (y < NP) {
            float Mf = M, Sf = S, zlab;
            if (labBad[i]) {
                // column y was -1 in the base pass but must be +1 (bad_pos)
                float zn = OIMS;                       // 30 * 1.0
                float Mp = fmaxf(Mf, zn);
                Sf = Sf * __expf(Mf - Mp) - __expf(-OIMS - Mp) + __expf(zn - Mp);
                Mf = Mp;
                zlab = zn;
            } else {
                zlab = OIMS * labVal[i];
            }
            float logp = zlab - Mf - __logf(Sf);
            float ii = iou[i];
            out = ii * ii * (-logp);                   // (-iou*logp)*iou
        }
        rowLoss[i] = out;
    }
}

// ---------------------------------------------------------------------------
// K4: reduce per-row losses -> d_out[0]
// ---------------------------------------------------------------------------
__global__ __launch_bounds__(256) void loss_sum(const float* __restrict__ rowLoss,
                                                const int* __restrict__ label,
                                                float* __restrict__ out0) {
    __shared__ float rs[256];
    __shared__ int   rc[256];
    const int t = threadIdx.x;
    float s = 0.f;
    int c = 0;
    for (int i = t; i < NROWS; i += 256) {
        s += rowLoss[i];
        c += (label[i] < NP) ? 1 : 0;
    }
    rs[t] = s; rc[t] = c;
    __syncthreads();
    for (int st = 128; st > 0; st >>= 1) {
        if (t < st) { rs[t] += rs[t + st]; rc[t] += rc[t + st]; }
        __syncthreads();
    }
    if (t == 0) out0[0] = rs[0] / (float)rc[0];
}

// ---------------------------------------------------------------------------
// K5: copy banks into output (lut then cq, contiguous after the loss scalar)
// ---------------------------------------------------------------------------
__global__ __launch_bounds__(256) void copy_banks(const float* __restrict__ lut,
                                                  const float* __restrict__ cq,
                                                  float* __restrict__ out) {
    const size_t lutN  = (size_t)NP * NF;
    const size_t total = lutN + (size_t)NQ * NF;
    for (size_t idx = (size_t)blockIdx.x * blockDim.x + threadIdx.x; idx < total;
         idx += (size_t)gridDim.x * blockDim.x)
        out[idx] = (idx < lutN) ? lut[idx] : cq[idx - lutN];
}

// ---------------------------------------------------------------------------
// K6: sequential (order-dependent) bank update. One block, 1 thread/feature.
// ---------------------------------------------------------------------------
__global__ __launch_bounds__(256) void update_banks(const int* __restrict__ label,
                                                    const float* __restrict__ iou,
                                                    const float* __restrict__ xn,
                                                    float* __restrict__ outLut,
                                                    float* __restrict__ outCq) {
    __shared__ float red[8];
    const int t = threadIdx.x;
    int head = 0;
    for (int i = 0; i < NROWS; ++i) {
        const int y = label[i];            // uniform across the block
        const float xi = xn[(size_t)i * NF + t];
        if (y < NP) {
            float* row = outLut + (size_t)y * NF;
            float old = row[t];
            float ii = iou[i];
            float nv = 0.5f * old + 0.5f * xi * ii;
            float ss = nv * nv;
            ss += __shfl_xor(ss, 1, 32);
            ss += __shfl_xor(ss, 2, 32);
            ss += __shfl_xor(ss, 4, 32);
            ss += __shfl_xor(ss, 8, 32);
            ss += __shfl_xor(ss, 16, 32);
            if ((t & 31) == 0) red[t >> 5] = ss;
            __syncthreads();
            float tot = 0.f;
#pragma unroll
            for (int k = 0; k < 8; ++k) tot += red[k];
            float n = sqrtf(tot);
            row[t] = nv / fmaxf(n, EPSN);
            __syncthreads();               // make update visible for repeats of y
        } else {
            outCq[(size_t)head * NF + t] = xi;
            ++head;
        }
    }
}

// ---------------------------------------------------------------------------
extern "C" void kernel_launch(void* const* d_in, const int* in_sizes, int n_in,
                              void* d_out, int out_size, void* d_ws, size_t ws_size,
                              hipStream_t stream) {
    (void)in_sizes; (void)n_in; (void)out_size; (void)ws_size;
    const float* inputs = (const float*)d_in[0];
    const int*   label  = (const int*)d_in[1];
    const float* iou    = (const float*)d_in[2];
    const float* lut    = (const float*)d_in[3];
    const float* cq     = (const float*)d_in[4];
    float* out = (float*)d_out;

    // workspace carve-out (~3.2 MB)
    float* xn      = (float*)d_ws;                       // 1024*256
    float* partM   = xn + (size_t)NROWS * NF;            // 1024*260
    float* partS   = partM + (size_t)NROWS * NBM_OUT;    // 1024*260
    float* labVal  = partS + (size_t)NROWS * NBM_OUT;    // 1024
    float* rowLoss = labVal + NROWS;                     // 1024
    int*   labBad  = (int*)(rowLoss + NROWS);            // 1024

    float* outLut = out + 1;
    float* outCq  = outLut + (size_t)NP * NF;

    normalize_rows<<<NROWS, 256, 0, stream>>>(inputs, xn);
    copy_banks<<<4096, 256, 0, stream>>>(lut, cq, outLut);
    oim_gemm_softmax<<<dim3(NBM_OUT, NROWS / 64), 256, 0, stream>>>(xn, lut, cq,
                                                                    partM, partS);
    label_logit<<<NROWS, 256, 0, stream>>>(label, xn, lut, labVal, labBad);
    row_finalize<<<NROWS, 256, 0, stream>>>(partM, partS, label, iou, labVal,
                                            labBad, rowLoss);
    loss_sum<<<1, 256, 0, stream>>>(rowLoss, label, out);
    update_banks<<<1, 256, 0, stream>>>(label, iou, xn, outLut, outCq);
}